// EntropyProfileLoss_60129542144534
// MI455X (gfx1250) — compile-verified
//
#include <hip/hip_runtime.h>
#include <hip/hip_bf16.h>
#include <math.h>

#define L          2048
#define ROWS       128
#define SEGS       4            // window segments per row
#define SEGW       512          // window positions per segment
#define NT         256
#define NWAVES     (NT / 32)    // 8 wave32s
#define NK         6
#define SLICE_POS  768          // local positions where s4 is computed (48 chunks of 16)
#define NCH        (SLICE_POS / 16)          // 48 WMMA chunks per array
#define ELEMS      (SLICE_POS + 4)           // slice elements (+pad for B reads to pos+18)

typedef __attribute__((ext_vector_type(2))) float v2f;
typedef __attribute__((ext_vector_type(8))) float v8f;

__device__ __forceinline__ float wave_max32(float v) {
    #pragma unroll
    for (int off = 16; off > 0; off >>= 1)
        v = fmaxf(v, __shfl_xor(v, off, 32));
    return v;
}

__global__ void __launch_bounds__(NT)
entropy_seg_kernel(const float* __restrict__ in,
                   const float* __restrict__ tg,
                   float* __restrict__ partials) {
    // e-buffers double as raw-x buffers before the transform.
    __shared__ float eA[ELEMS], fA[ELEMS], eB[ELEMS], fB[ELEMS];
    __shared__ float sAe[SLICE_POS], sAf[SLICE_POS], sBe[SLICE_POS], sBf[SLICE_POS];
    __shared__ float redA[NWAVES], redB[NWAVES];
    __shared__ float redT[NT];

    const int tid  = (int)threadIdx.x;
    const int lane = tid & 31;
    const int wv   = tid >> 5;
    const int row  = (int)blockIdx.x >> 2;      // SEGS == 4
    const int seg  = (int)blockIdx.x & 3;
    const int base = seg * SEGW;                // first window position of this segment

    const float* __restrict__ xin = in + (size_t)row * L;
    const float* __restrict__ xtg = tg + (size_t)row * L;

    // ---- Phase 1: load slice [base, base+ELEMS) into LDS, zero past row end ----
    for (int i = tid; i < ELEMS; i += NT) {
        int g = base + i;
        bool ok = (g < L);
        eA[i] = ok ? xin[g] : 0.0f;
        eB[i] = ok ? xtg[g] : 0.0f;
    }
    __syncthreads();

    // ---- Phase 2: slice max (any shift is valid; max keeps exp <= 1) ----
    float mA = -INFINITY, mB = -INFINITY;
    for (int i = tid; i < ELEMS; i += NT) {
        mA = fmaxf(mA, eA[i]);
        mB = fmaxf(mB, eB[i]);
    }
    mA = wave_max32(mA);
    mB = wave_max32(mB);
    if (lane == 0) { redA[wv] = mA; redB[wv] = mB; }
    __syncthreads();
    if (tid == 0) {
        float a = redA[0], b = redB[0];
        #pragma unroll
        for (int w = 1; w < NWAVES; ++w) { a = fmaxf(a, redA[w]); b = fmaxf(b, redB[w]); }
        redA[0] = a; redB[0] = b;
    }
    __syncthreads();
    mA = redA[0];
    mB = redB[0];

    // ---- Phase 3: e = exp(x-M), f = e*(x-M), in place ----
    for (int i = tid; i < ELEMS; i += NT) {
        float xa = eA[i] - mA;
        float ea = __expf(xa);
        eA[i] = ea; fA[i] = ea * xa;
        float xb = eB[i] - mB;
        float eb = __expf(xb);
        eB[i] = eb; fB[i] = eb * xb;
    }
    __syncthreads();

    // ---- Phase 4: k=4 sliding-window sums via V_WMMA_F32_16X16X4_F32 ----
    // D = A(16x4 ones) x B(4x16), B[k][n] = src[s0+n+k]  =>  D[m,n] = s4[s0+n] for all m.
    // 2-chunk unroll: lanes 0-15 store chunk c's row, lanes 16-31 store chunk (c+8)'s row
    // (branchless cndmask select; EXEC stays all-ones through the WMMAs).
    {
        v2f a_ones; a_ones[0] = 1.0f; a_ones[1] = 1.0f;
        const int n    = lane & 15;
        const int k0   = lane >> 4;
        const bool lo  = (lane < 16);
        const float* srcs[4] = { eA, fA, eB, fB };
        float*       dsts[4] = { sAe, sAf, sBe, sBf };
        #pragma unroll
        for (int arr = 0; arr < 4; ++arr) {
            const float* __restrict__ s = srcs[arr];
            float* __restrict__ d = dsts[arr];
            #pragma unroll
            for (int p = 0; p < NCH / (2 * NWAVES); ++p) {   // 3 pairs per wave
                const int s0 = (wv + p * 2 * NWAVES) * 16;
                const int s1 = s0 + NWAVES * 16;
                v2f b0, b1;
                b0[0] = s[s0 + n + k0];
                b0[1] = s[s0 + n + k0 + 2];
                b1[0] = s[s1 + n + k0];
                b1[1] = s[s1 + n + k0 + 2];
                v8f z = {};
#if __has_builtin(__builtin_amdgcn_wmma_f32_16x16x4_f32)
                v8f acc0 = __builtin_amdgcn_wmma_f32_16x16x4_f32(
                               false, a_ones, false, b0, (short)0, z, false, false);
                v8f acc1 = __builtin_amdgcn_wmma_f32_16x16x4_f32(
                               false, a_ones, false, b1, (short)0, z, false, false);
#else
                v8f acc0 = z, acc1 = z;
                acc0[0] = s[s0 + n] + s[s0 + n + 1] + s[s0 + n + 2] + s[s0 + n + 3];
                acc1[0] = s[s1 + n] + s[s1 + n + 1] + s[s1 + n + 2] + s[s1 + n + 3];
#endif
                const int   addr = lo ? (s0 + n) : (s1 + n);
                const float val  = lo ? acc0[0] : acc1[0];
                d[addr] = val;
            }
        }
    }
    __syncthreads();

    // ---- Phase 5: per-k NE |diff| accumulation + in-place doubling ----
    const int ks[NK] = {4, 8, 16, 32, 64, 128};
    float total = 0.0f;
    for (int j = 0; j < NK; ++j) {
        const int k = ks[j];
        int nwin = L - k + 1 - base;            // windows this segment owns
        if (nwin > SEGW) nwin = SEGW;
        float accj = 0.0f;
        for (int i = tid; i < nwin; i += NT) {  // at most 2 trips
            float SeA = sAe[i], SfA = sAf[i];
            float neA = SfA * __builtin_amdgcn_rcpf(SeA) - __logf(SeA);
            float SeB = sBe[i], SfB = sBf[i];
            float neB = SfB * __builtin_amdgcn_rcpf(SeB) - __logf(SeB);
            accj += fabsf(neA - neB);
        }
        total += accj * (1.0f / (128.0f * (float)(L - k + 1)));

        if (j + 1 < NK) {
            // s_{2k}[i] = s_k[i] + s_k[i+k], valid while i+k < SLICE_POS
            const int lim = SLICE_POS - 1 - k;  // inclusive
            float rAe[3], rAf[3], rBe[3], rBf[3];
            #pragma unroll
            for (int t = 0; t < 3; ++t) {       // SLICE_POS / NT == 3
                int i = tid + t * NT;
                if (i <= lim) {
                    rAe[t] = sAe[i] + sAe[i + k];
                    rAf[t] = sAf[i] + sAf[i + k];
                    rBe[t] = sBe[i] + sBe[i + k];
                    rBf[t] = sBf[i] + sBf[i + k];
                } else {
                    rAe[t] = 0.f; rAf[t] = 0.f; rBe[t] = 0.f; rBf[t] = 0.f;
                }
            }
            __syncthreads();                    // all reads before any write
            #pragma unroll
            for (int t = 0; t < 3; ++t) {
                int i = tid + t * NT;
                if (i <= lim) {
                    sAe[i] = rAe[t]; sAf[i] = rAf[t];
                    sBe[i] = rBe[t]; sBf[i] = rBf[t];
                }
            }
            __syncthreads();                    // writes visible for next k
        }
    }

    // ---- Phase 6: deterministic block reduction -> per-segment partial ----
    redT[tid] = total;
    __syncthreads();
    #pragma unroll
    for (int s = NT / 2; s > 0; s >>= 1) {
        if (tid < s) redT[tid] += redT[tid + s];
        __syncthreads();
    }
    if (tid == 0) partials[blockIdx.x] = redT[0];
}

__global__ void finalize_kernel(const float* __restrict__ partials,
                                float* __restrict__ out) {
    if (threadIdx.x == 0 && blockIdx.x == 0) {
        float s = 0.0f;
        for (int i = 0; i < ROWS * SEGS; ++i) s += partials[i];  // fixed order
        out[0] = s;
    }
}

extern "C" void kernel_launch(void* const* d_in, const int* in_sizes, int n_in,
                              void* d_out, int out_size, void* d_ws, size_t ws_size,
                              hipStream_t stream) {
    (void)in_sizes; (void)n_in; (void)out_size; (void)ws_size;
    const float* in = (const float*)d_in[0];
    const float* tg = (const float*)d_in[1];
    float* out      = (float*)d_out;
    float* partials = (float*)d_ws;   // ROWS*SEGS = 512 floats of scratch

    entropy_seg_kernel<<<ROWS * SEGS, NT, 0, stream>>>(in, tg, partials);
    finalize_kernel<<<1, 32, 0, stream>>>(partials, out);
}